// NF_20581483282566
// MI455X (gfx1250) — compile-verified
//
#include <hip/hip_runtime.h>
#include <hip/hip_bf16.h>

typedef __attribute__((ext_vector_type(2))) float v2f;
typedef __attribute__((ext_vector_type(8))) float v8f;

// ELU with alpha=1 (jax.nn.elu default)
__device__ __forceinline__ float elu1(float v) {
    return v > 0.0f ? v : expm1f(v);
}

// Stage 1+2: streaming pass over x (102.4 MB -> bandwidth bound).
// One thread per sample b: reads its contiguous 512B block via float4,
// computes 4 strided-conv outputs + ELU, then the 1x4 conv + ELU,
// writes (sr, si) into the padded workspace stream.
__global__ void nf_stage12_kernel(const float* __restrict__ x,
                                  const float* __restrict__ w1,   // (1,2,16) flat: [0..15]=ch0, [16..31]=ch1
                                  const float* __restrict__ b1,   // (1,)
                                  const float* __restrict__ w2,   // (2,1,4) flat: [0..3]=out0, [4..7]=out1
                                  const float* __restrict__ b2,   // (2,)
                                  float* __restrict__ srp,        // padded: [0..15]=0, [16+b]=sr[b]
                                  float* __restrict__ sip,        // padded likewise
                                  int Bn) {
    int b = blockIdx.x * blockDim.x + threadIdx.x;
    if (b < 16) {           // zero the 16-element causal pads every call (deterministic)
        srp[b] = 0.0f;
        sip[b] = 0.0f;
    }
    if (b >= Bn) return;

    const float* xb = x + (size_t)b * 128;   // [0..63]=ch0, [64..127]=ch1
    float bias1 = b1[0];

    float h[4];
#pragma unroll
    for (int o = 0; o < 4; ++o) {
        float acc = bias1;
#pragma unroll
        for (int kk = 0; kk < 4; ++kk) {
            float4 v0 = *(const float4*)(xb + o * 16 + kk * 4);        // channel 0
            float4 v1 = *(const float4*)(xb + 64 + o * 16 + kk * 4);   // channel 1
            acc = fmaf(v0.x, w1[kk * 4 + 0], acc);
            acc = fmaf(v0.y, w1[kk * 4 + 1], acc);
            acc = fmaf(v0.z, w1[kk * 4 + 2], acc);
            acc = fmaf(v0.w, w1[kk * 4 + 3], acc);
            acc = fmaf(v1.x, w1[16 + kk * 4 + 0], acc);
            acc = fmaf(v1.y, w1[16 + kk * 4 + 1], acc);
            acc = fmaf(v1.z, w1[16 + kk * 4 + 2], acc);
            acc = fmaf(v1.w, w1[16 + kk * 4 + 3], acc);
        }
        h[o] = elu1(acc);
    }

    float r0 = b2[0], r1 = b2[1];
#pragma unroll
    for (int k = 0; k < 4; ++k) {
        r0 = fmaf(h[k], w2[k], r0);
        r1 = fmaf(h[k], w2[4 + k], r1);
    }
    srp[16 + b] = elu1(r0);
    sip[16 + b] = elu1(r1);
}

// Stage 3: 16-tap causal complex FIR across the batch axis, mapped onto
// V_WMMA_F32_16X16X4_F32 (full fp32 — matches reference numerics).
// One wave32 handles 16 consecutive outputs (M dim). K enumerates
// (tap j, component c) pairs: K = 2j + c, 32 total -> 8 chained K=4 WMMAs.
// B-matrix column n=0 holds (tr[j], -ti[j]); n=1 holds (ti[j], tr[j]);
// columns 2..15 are zero. D[:,0] = y_real, D[:,1] = y_imag.
__global__ void nf_fir_wmma_kernel(const float* __restrict__ srp,
                                   const float* __restrict__ sip,
                                   const float* __restrict__ tr,
                                   const float* __restrict__ ti,
                                   float* __restrict__ out,   // (B,2,1) flat: out[2b]=re, out[2b+1]=im
                                   int Bn) {
    int gtid = blockIdx.x * blockDim.x + threadIdx.x;
    int wave = gtid >> 5;            // blockDim multiple of 32
    int lane = threadIdx.x & 31;
    int b0 = wave * 16;              // wave-uniform -> uniform branch keeps EXEC all-1s
    if (b0 >= Bn) return;

    int n    = lane & 15;            // A: M row = lane%16; B/D: N col = lane%16
    int half = lane >> 4;            // half-wave selects K pair / M half

    v8f acc = {};
#pragma unroll
    for (int q = 0; q < 8; ++q) {
        // This WMMA's local K=0..3 covers global K = 4q..4q+3.
        // Lane supplies A[m, 4q+2*half] and A[m, 4q+2*half+1]  (even k -> sr, odd k -> si)
        int j   = 2 * q + half;              // tap index for this lane's K pair
        int idx = 16 + b0 + n - j;           // padded stream index, always >= 1
        v2f a;
        a.x = srp[idx];                      // A[m, k]   (c=0 -> sr[b0+m-j])
        a.y = sip[idx];                      // A[m, k+1] (c=1 -> si[b0+m-j])

        float trj = tr[j];
        float tij = ti[j];
        v2f bb;                              // B[k, n], B[k+1, n]
        bb.x = (n == 0) ? trj : ((n == 1) ? tij : 0.0f);
        bb.y = (n == 0) ? -tij : ((n == 1) ? trj : 0.0f);

        acc = __builtin_amdgcn_wmma_f32_16x16x4_f32(
            /*neg_a=*/false, a, /*neg_b=*/false, bb,
            /*c_mod=*/(short)0, acc, /*reuse_a=*/false, /*reuse_b=*/false);
    }

    // D layout: VGPR r holds M = r + 8*(lane/16), N = lane%16. Only N<2 carry data.
    if (n < 2) {
#pragma unroll
        for (int r = 0; r < 8; ++r) {
            int b = b0 + half * 8 + r;
            if (b < Bn) out[2 * b + n] = acc[r];
        }
    }
}

extern "C" void kernel_launch(void* const* d_in, const int* in_sizes, int n_in,
                              void* d_out, int out_size, void* d_ws, size_t ws_size,
                              hipStream_t stream) {
    const float* x  = (const float*)d_in[0];
    const float* w1 = (const float*)d_in[1];
    const float* b1 = (const float*)d_in[2];
    const float* w2 = (const float*)d_in[3];
    const float* b2 = (const float*)d_in[4];
    const float* tr = (const float*)d_in[5];
    const float* ti = (const float*)d_in[6];
    float* out = (float*)d_out;

    int Bn = in_sizes[0] / 128;                    // (B, 2, 64) -> B
    int spad = 16 + ((Bn + 15) & ~15);             // 16-elem causal pad + tile-rounded body
    float* srp = (float*)d_ws;                     // needs 2*spad*4 bytes (~1.6 MB)
    float* sip = srp + spad;

    dim3 blk(256);
    dim3 g1((Bn + 255) / 256);
    nf_stage12_kernel<<<g1, blk, 0, stream>>>(x, w1, b1, w2, b2, srp, sip, Bn);

    int waves   = (Bn + 15) / 16;                  // one wave32 per 16-sample tile
    int threads = waves * 32;
    dim3 g2((threads + 255) / 256);
    nf_fir_wmma_kernel<<<g2, blk, 0, stream>>>(srp, sip, tr, ti, out, Bn);
}